// OctreeTrilinear_8624294331040
// MI455X (gfx1250) — compile-verified
//
#include <hip/hip_runtime.h>
#include <stdint.h>

typedef unsigned int u32x4 __attribute__((ext_vector_type(4)));
typedef int          i32x4 __attribute__((ext_vector_type(4)));
typedef int          i32x8 __attribute__((ext_vector_type(8)));
typedef float        v2f   __attribute__((ext_vector_type(2)));
typedef float        v8f   __attribute__((ext_vector_type(8)));

#define C_CH   32
#define GRID   64
#define NPTS   128      // points per workgroup
#define TPB    256      // 8 wave32
#define PITCH  132      // LDS row pitch (floats): multiple of 4 -> b128-aligned

#if defined(__has_builtin)
#if __has_builtin(__builtin_amdgcn_wmma_f32_16x16x4_f32)
#define HAVE_WMMA4 1
#endif
#endif

// ---------------------------------------------------------------------------
// Kernel 1a (matrix-pipe path): transpose (C,H) -> (H,C) via WMMA x identity.
// A[m,k] = src[cb+m][h+4i+k]  (A-layout: row per lane, K striped in VGPRs)
// B_i[k,n] = delta(n, 4i+k)   (shifted identity)
// After 4 accumulating WMMAs: D[m,n] = src[cb+m][h+n]  (D-layout: row across
// lanes) -> 2x b128 stores per lane into the (H,C) destination. No LDS.
// Each wave: one 16c x 16h tile; block = 8 waves = 32c x 64h.
// ---------------------------------------------------------------------------
#ifdef HAVE_WMMA4
__global__ __launch_bounds__(TPB) void octt_transpose_wmma(const float* __restrict__ src,
                                                           float* __restrict__ dst,
                                                           int H) {
    const int t  = threadIdx.x;
    const int l  = t & 31;
    const int w  = t >> 5;
    const int n  = l & 15;        // D column (h offset) / A row (channel)
    const int hi = l >> 4;        // lane half -> K pair {0,1} or {2,3}
    const int cb = (w & 1) * 16;  // channel base
    const int h  = blockIdx.x * 64 + (w >> 1) * 16;   // h window [h, h+16)

    const float* rowp = src + (size_t)(cb + n) * H + h + 2 * hi;

    v8f d = { 0.f, 0.f, 0.f, 0.f, 0.f, 0.f, 0.f, 0.f };
    #pragma unroll
    for (int i = 0; i < 4; ++i) {
        v2f a;                                  // b64 load: K pair of row cb+n
        a.x = rowp[4 * i + 0];
        a.y = rowp[4 * i + 1];
        v2f b;                                  // shifted identity column
        b.x = (n == 4 * i + 2 * hi    ) ? 1.0f : 0.0f;
        b.y = (n == 4 * i + 2 * hi + 1) ? 1.0f : 0.0f;
        d = __builtin_amdgcn_wmma_f32_16x16x4_f32(false, a, false, b,
                                                  (short)0, d, false, false);
    }

    // D[m,n]: VGPR r -> m = r + 8*hi, lane column n -> h+n
    float* op = dst + (size_t)(h + n) * C_CH + cb + 8 * hi;
    *(float4*)(op)     = make_float4(d[0], d[1], d[2], d[3]);
    *(float4*)(op + 4) = make_float4(d[4], d[5], d[6], d[7]);
}
#endif

// ---------------------------------------------------------------------------
// Kernel 1b (fallback, H % 64 != 0): LDS tile transpose.
// ---------------------------------------------------------------------------
__global__ __launch_bounds__(TPB) void octt_transpose(const float* __restrict__ src,
                                                      float* __restrict__ dst,
                                                      int H) {
    __shared__ float tile[C_CH][65];
    const int t  = threadIdx.x;
    const int h0 = blockIdx.x * 64;

    const float* sp = src + (size_t)(t >> 6) * H + h0 + (t & 63);
    #pragma unroll
    for (int r = 0; r < 8; ++r) {              // coalesced reads along h
        int c = r * 4 + (t >> 6);
        int i = t & 63;
        int h = h0 + i;
        tile[c][i] = (h < H) ? sp[0] : 0.0f;
        sp += (size_t)4 * H;                   // pointer bump: no per-row mul64
    }
    __syncthreads();
    float* dp = dst + (size_t)(h0 + (t >> 5)) * C_CH + (t & 31);
    #pragma unroll
    for (int r = 0; r < 8; ++r) {              // coalesced writes along c
        int h = h0 + r * 8 + (t >> 5);
        if (h < H) dp[0] = tile[t & 31][r * 8 + (t >> 5)];
        dp += 8 * C_CH;
    }
}

// ---------------------------------------------------------------------------
// Kernel 2: trilinear gather. wave-per-point, lane-per-channel.
// pts tile staged into LDS by the Tensor Data Mover (async, TENSORcnt).
// ---------------------------------------------------------------------------
__global__ __launch_bounds__(TPB) void octt_trilerp(const float* __restrict__ feat,
                                                    const float* __restrict__ pts,
                                                    const int*   __restrict__ lut,
                                                    float*       __restrict__ out,
                                                    int N, int H, int transposed) {
    __shared__ float s_pts[NPTS * 4];                       // TDM destination
    __shared__ __align__(16) float s_tr[C_CH * PITCH];      // padded transpose buffer

    const int t    = threadIdx.x;
    const int lane = t & 31;                       // channel
    const int wave = t >> 5;
    const int base = blockIdx.x * NPTS;
    const int npts = (N - base < NPTS) ? (N - base) : NPTS;

    // ---- TDM: stage this block's point tile (wave 0 issues once) ----------
    if (t < 32) {
        uint64_t ga   = (uint64_t)(uintptr_t)(pts + (size_t)base * 4);
        uint32_t ldso = (uint32_t)(uintptr_t)(&s_pts[0]);       // LDS byte offset
        uint32_t remf = (uint32_t)((uint64_t)(N - base) * 4u);  // floats left -> OOB clamp

        // D# group0: count=1 | lds_addr | global_addr[56:0] | type=2
        u32x4 g0 = { 1u,
                     ldso,
                     (uint32_t)ga,
                     (uint32_t)((ga >> 32) & 0x01FFFFFFu) | (2u << 30) };

        // D# group1: data_size=4B, 1-D tile of NPTS*4 floats,
        // tensor_dim0 = remaining floats (tail reads return zero), dim1=1.
        uint32_t dim0 = remf;
        uint32_t tl0  = NPTS * 4u;                  // 512 elements
        i32x8 g1 = { (int)(2u << 16),                               // wgmask=0, data_size=2(4B)
                     (int)((dim0 & 0xFFFFu) << 16),                 // tensor_dim0[15:0]
                     (int)((dim0 >> 16) | (1u << 16)),              // dim0[31:16] | tensor_dim1=1
                     (int)(tl0 << 16),                              // dim1[31:16]=0 | tile_dim0
                     0,                                             // tile_dim1=0, tile_dim2=0
                     (int)tl0,                                      // tensor_dim0_stride
                     0, 0 };                                        // stride hi, dim1_stride
        i32x4 g2 = { 0, 0, 0, 0 };
        i32x4 g3 = { 0, 0, 0, 0 };
        i32x8 g4 = { 0, 0, 0, 0, 0, 0, 0, 0 };     // extra group (clang-23 6-arg form)

        __builtin_amdgcn_tensor_load_to_lds(g0, g1, g2, g3, g4, 0);
        __builtin_amdgcn_s_wait_tensorcnt(0);
    }
    __syncthreads();

    // ---- per-point gather + weighted accumulate ---------------------------
    for (int j = 0; j < NPTS / 8; ++j) {           // 16 points per wave
        int p = wave * (NPTS / 8) + j;
        if (p >= npts) break;

        float px = s_pts[p * 4 + 0] - 0.5f;        // LDS broadcast (wave-uniform)
        float py = s_pts[p * 4 + 1] - 0.5f;
        float pz = s_pts[p * 4 + 2] - 0.5f;
        float flx = floorf(px), fly = floorf(py), flz = floorf(pz);
        float fx = px - flx, fy = py - fly, fz = pz - flz;
        int ix = (int)flx, iy = (int)fly, iz = (int)flz;

        float acc = 0.0f, wsum = 0.0f;
        #pragma unroll
        for (int k = 0; k < 8; ++k) {
            const int mx = (k >> 2) & 1, my = (k >> 1) & 1, mz = k & 1;
            int cx = ix + mx, cy = iy + my, cz = iz + mz;
            bool inb = ((unsigned)cx < (unsigned)GRID) &
                       ((unsigned)cy < (unsigned)GRID) &
                       ((unsigned)cz < (unsigned)GRID);
            float w = fabsf(((1.0f - (float)mx) - fx) *
                            ((1.0f - (float)my) - fy) *
                            ((1.0f - (float)mz) - fz));
            int idx = -1;
            if (inb) {
                // wave-uniform -> force SGPR so the LUT read scalarizes (s_load)
                int flat = __builtin_amdgcn_readfirstlane((cx << 12) | (cy << 6) | cz);
                idx = lut[flat];
            }
            if (idx > -1) {                         // wave-uniform branch
                int is = __builtin_amdgcn_readfirstlane(idx);
                float f = transposed ? feat[(size_t)is * C_CH + lane]   // 128B coalesced
                                     : feat[(size_t)lane * H + is];     // fallback (no ws)
                acc  += w * f;
                wsum += w;
            }
        }
        // v_rcp_f32 instead of IEEE div: wave-uniform scale, eps-guarded anyway
        s_tr[lane * PITCH + p] = acc * __builtin_amdgcn_rcpf(wsum + 1e-10f);
    }
    __syncthreads();

    // ---- coalesced store: out[c*N + base + p], b128-vectorized ------------
    if (((N | base) & 3) == 0) {
        // 32 rows x 32 float4 = 1024 float4s; 4 passes of 256 threads
        #pragma unroll
        for (int r = 0; r < (C_CH * NPTS / 4) / TPB; ++r) {
            int f  = t + r * TPB;
            int c  = f >> 5;                        // /(NPTS/4)
            int p4 = f & 31;
            int p  = p4 * 4;
            if (p + 3 < npts) {
                float4 v = *(const float4*)&s_tr[c * PITCH + p];
                *(float4*)&out[(size_t)c * N + base + p] = v;
            } else if (p < npts) {
                for (int e = 0; e < 4 && p + e < npts; ++e)
                    out[(size_t)c * N + base + p + e] = s_tr[c * PITCH + p + e];
            }
        }
    } else {                                        // unaligned fallback
        #pragma unroll
        for (int r = 0; r < (C_CH * NPTS) / TPB; ++r) {
            int f = t + r * TPB;
            int c = f >> 7;
            int p = f & (NPTS - 1);
            if (p < npts) out[(size_t)c * N + base + p] = s_tr[c * PITCH + p];
        }
    }
}

// ---------------------------------------------------------------------------
extern "C" void kernel_launch(void* const* d_in, const int* in_sizes, int n_in,
                              void* d_out, int out_size, void* d_ws, size_t ws_size,
                              hipStream_t stream) {
    const float* data = (const float*)d_in[0];   // (1,C,H,1) -> data[c*H+h]
    const float* pts  = (const float*)d_in[1];   // (N,4)
    const int*   lut  = (const int*)d_in[2];     // (64,64,64)
    // d_in[3] = depth (6) -> G=64, compile-time

    const int N = in_sizes[1] / 4;
    const int H = in_sizes[0] / C_CH;

    float* feat = (float*)d_ws;
    const int use_t = (ws_size >= (size_t)H * C_CH * sizeof(float)) ? 1 : 0;

    if (use_t) {
#ifdef HAVE_WMMA4
        if ((H & 63) == 0) {
            octt_transpose_wmma<<<H / 64, TPB, 0, stream>>>(data, feat, H);
        } else
#endif
        {
            octt_transpose<<<(H + 63) / 64, TPB, 0, stream>>>(data, feat, H);
        }
    }
    octt_trilerp<<<(N + NPTS - 1) / NPTS, TPB, 0, stream>>>(
        use_t ? feat : data, pts, lut, (float*)d_out, N, H, use_t);
}